// HierarchicalBlockShuffleLayer_66279935311935
// MI455X (gfx1250) — compile-verified
//
#include <hip/hip_runtime.h>

// ---------------- problem constants (from reference) ----------------
#define EPSV 1e-5f
constexpr int D    = 1024;
constexpr int SQ   = 32;    // monarch grid: 32x32
constexpr int NG   = 64;    // block-diag groups
constexpr int TILE = 16;    // rows per workgroup (WMMA M)

// LDS pitches (floats), chosen for bank-conflict-free WMMA fragment access
constexpr int SUBP = 36;            // pitch of one 32-col sub-row
constexpr int ROWP = SQ * SUBP + 4; // 1156: pitch of one full 1024-col row
constexpr int TP   = 36;            // pitch of one 32-col T row

typedef __attribute__((ext_vector_type(2))) float v2f;
typedef __attribute__((ext_vector_type(8))) float v8f;

__device__ __forceinline__ int xidx(int row, int d) {
    return row * ROWP + (d >> 5) * SUBP + (d & 31);
}

__global__ __launch_bounds__(256, 1)
void hbs_fused_kernel(const float* __restrict__ x,
                      const float* __restrict__ gamma,
                      const float* __restrict__ beta,
                      const float* __restrict__ bw,     // [64][16][16]
                      const float* __restrict__ Wrow,   // [32][32]
                      const float* __restrict__ Wcol,   // [32][32]
                      const float* __restrict__ alpha_p,
                      const float* __restrict__ bias,
                      float* __restrict__ out)
{
    __shared__ float xs [TILE * ROWP];        // layernormed tile
    __shared__ float acc[TILE * ROWP];        // output accumulator
    __shared__ float Ts [TILE * SQ * TP];     // monarch intermediate (512 x 32)

    const int tid   = threadIdx.x;
    const int lane  = tid & 31;
    const int wave  = tid >> 5;
    const int tile0 = blockIdx.x * TILE;
    const float alpha = alpha_p[0];

    // ---------------- Phase 0: LayerNorm (2 rows per wave) ----------------
    for (int rr = wave * 2; rr < wave * 2 + 2; ++rr) {
        const float4* xr = (const float4*)(x + (size_t)(tile0 + rr) * D);
        float4 v[8];
        float s = 0.f, s2 = 0.f;
#pragma unroll
        for (int i = 0; i < 8; ++i) {
            v[i] = xr[i * 32 + lane];
            s  += v[i].x + v[i].y + v[i].z + v[i].w;
            s2 += v[i].x * v[i].x + v[i].y * v[i].y + v[i].z * v[i].z + v[i].w * v[i].w;
        }
#pragma unroll
        for (int off = 16; off > 0; off >>= 1) {
            s  += __shfl_xor(s,  off, 32);
            s2 += __shfl_xor(s2, off, 32);
        }
        const float mu   = s * (1.f / D);
        const float var  = s2 * (1.f / D) - mu * mu;
        const float rstd = rsqrtf(var + EPSV);
#pragma unroll
        for (int i = 0; i < 8; ++i) {
            const int e  = i * 32 + lane;      // float4 index within row
            float4 gv = ((const float4*)gamma)[e];
            float4 bv = ((const float4*)beta)[e];
            float4 o;
            o.x = (v[i].x - mu) * rstd * gv.x + bv.x;
            o.y = (v[i].y - mu) * rstd * gv.y + bv.y;
            o.z = (v[i].z - mu) * rstd * gv.z + bv.z;
            o.w = (v[i].w - mu) * rstd * gv.w + bv.w;
            *(float4*)(xs + xidx(rr, e * 4)) = o;
        }
    }
    __syncthreads();   // xs ready

    const int am    = lane & 15;          // M (or N) index within half-wave
    const int khalf = (lane >> 4) << 1;   // K sub-pair: lanes 0-15 -> K{0,1}, 16-31 -> K{2,3}
    const int hi8   = (lane >> 4) << 3;   // D-matrix M offset for upper half-wave

    // ---------------- Phase 1: block-diagonal mix (+flip) ----------------
    for (int g = wave; g < NG; g += 8) {
        const float* Wg = bw + g * 256;   // [16][16] row-major (b,c)
        // B fragments for the 4 K-chunks (each wave's group is unique; L2-resident)
        v2f bf[4];
#pragma unroll
        for (int kk = 0; kk < 4; ++kk) {
            const int k0 = kk * 4 + khalf;
            bf[kk].x = Wg[(k0    ) * 16 + am];
            bf[kk].y = Wg[(k0 + 1) * 16 + am];
        }
        const float* arow = xs + xidx(am, g * 16);
        v8f c = {};
#pragma unroll
        for (int kk = 0; kk < 4; ++kk) {
            const int k0 = kk * 4 + khalf;
            v2f a;
            a.x = arow[k0];
            a.y = arow[k0 + 1];
            c = __builtin_amdgcn_wmma_f32_16x16x4_f32(false, a, false, bf[kk],
                                                      (short)0, c, false, false);
        }
        const int col = (NG - 1 - g) * 16 + (15 - am);   // flipped destination column
        float* dst = acc + xidx(hi8, col);
#pragma unroll
        for (int vv = 0; vv < 8; ++vv)
            dst[vv * ROWP] = c[vv];
    }

    // ------- Phases 2/3 share the wave-constant N-tile: nt = wave & 1 -------
    const int nt   = wave & 1;
    const int ncol = nt * 16 + am;        // output column index (k for GEMM1, s for GEMM2)
    const int mt0  = wave >> 1;           // first M-tile; stride 4 covers mt = 0..31

    // ---------------- Phase 2: monarch GEMM1  T = Xf @ W_col ----------------
    {
        // B fragments are loop-invariant for this wave: load once.
        v2f bf[8];
#pragma unroll
        for (int kk = 0; kk < 8; ++kk) {
            const int k0 = kk * 4 + khalf;           // c index
            bf[kk].x = Wcol[(k0    ) * 32 + ncol];
            bf[kk].y = Wcol[(k0 + 1) * 32 + ncol];
        }
        for (int mt = mt0; mt < 32; mt += 4) {
            const int f = mt * 16 + am;              // flattened row for A
            const float* arow = xs + (f >> 5) * ROWP + (f & 31) * SUBP;
            v8f c = {};
#pragma unroll
            for (int kk = 0; kk < 8; ++kk) {
                const int k0 = kk * 4 + khalf;
                v2f a;
                a.x = arow[k0];
                a.y = arow[k0 + 1];
                c = __builtin_amdgcn_wmma_f32_16x16x4_f32(false, a, false, bf[kk],
                                                          (short)0, c, false, false);
            }
            float* dst = Ts + (mt * 16 + hi8) * TP + ncol;
#pragma unroll
            for (int vv = 0; vv < 8; ++vv)
                dst[vv * TP] = c[vv];
        }
    }
    __syncthreads();   // Ts ready, acc (phase 1) complete everywhere

    // ------- Phase 3: monarch GEMM2  Z = T @ W_row^T, acc += alpha*Z -------
    {
        // B[k][s] = W_row[s][k]; row `ncol` of W_row is contiguous in k -> b64 loads
        const float* brow = Wrow + ncol * 32;
        v2f bf[8];
#pragma unroll
        for (int kk = 0; kk < 8; ++kk) {
            const int k0 = kk * 4 + khalf;
            bf[kk].x = brow[k0];
            bf[kk].y = brow[k0 + 1];
        }
        for (int mt = mt0; mt < 32; mt += 4) {
            const float* arow = Ts + (mt * 16 + am) * TP;
            v8f c = {};
#pragma unroll
            for (int kk = 0; kk < 8; ++kk) {
                const int k0 = kk * 4 + khalf;
                v2f a;
                a.x = arow[k0];
                a.y = arow[k0 + 1];
                c = __builtin_amdgcn_wmma_f32_16x16x4_f32(false, a, false, bf[kk],
                                                          (short)0, c, false, false);
            }
#pragma unroll
            for (int vv = 0; vv < 8; ++vv) {
                const int f  = mt * 16 + hi8 + vv;   // flattened row
                const int rr = f >> 5, r = f & 31;
                const int d  = r * 32 + ncol;        // unique owner per (f, s): no race
                acc[xidx(rr, d)] += alpha * c[vv];
            }
        }
    }
    __syncthreads();   // acc final

    // ---------------- Phase 4: writeback out = acc + bias ----------------
    // 16 rows * 256 float4 per row = 4096 float4, 256 threads -> 16 each
#pragma unroll
    for (int k = 0; k < 16; ++k) {
        const int j  = tid + k * 256;
        const int rr = j >> 8;
        const int cj = j & 255;          // float4 index within row
        float4 a = *(const float4*)(acc + xidx(rr, cj * 4));
        float4 b = ((const float4*)bias)[cj];
        float4 o;
        o.x = a.x + b.x; o.y = a.y + b.y; o.z = a.z + b.z; o.w = a.w + b.w;
        ((float4*)(out + (size_t)(tile0 + rr) * D))[cj] = o;
    }
}

extern "C" void kernel_launch(void* const* d_in, const int* in_sizes, int n_in,
                              void* d_out, int out_size, void* d_ws, size_t ws_size,
                              hipStream_t stream) {
    const float* x     = (const float*)d_in[0];
    const float* gamma = (const float*)d_in[1];
    const float* beta  = (const float*)d_in[2];
    const float* bw    = (const float*)d_in[3];
    const float* Wrow  = (const float*)d_in[4];
    const float* Wcol  = (const float*)d_in[5];
    const float* alpha = (const float*)d_in[6];
    const float* bias  = (const float*)d_in[7];
    float* out = (float*)d_out;

    const int nrows  = in_sizes[0] / D;        // B*S = 32768
    const int nblock = nrows / TILE;           // 2048 workgroups
    hbs_fused_kernel<<<nblock, 256, 0, stream>>>(x, gamma, beta, bw, Wrow, Wcol,
                                                 alpha, bias, out);
}